// SpectralConv1d_13108240187865
// MI455X (gfx1250) — compile-verified
//
#include <hip/hip_runtime.h>
#include <math.h>

typedef __attribute__((ext_vector_type(2))) float v2f;
typedef __attribute__((ext_vector_type(8))) float v8f;

#define B_SZ   32
#define CIN    32
#define COUT   32
#define MODES  64
#define LEN    16384
#define ROWS   (B_SZ * CIN)    // 1024 input rows  (b,i)
#define OROWS  (B_SZ * COUT)   // 1024 output rows (b,o)
#define KSPLIT 32
#define KCHUNK (LEN / KSPLIT)  // 512

// ---------------------------------------------------------------------------
// Kernel 1: cas basis table  T[k][n] = cos(2*pi*k*n/L) + sin(2*pi*k*n/L)
// (k*n mod L) is exact since L = 2^14  ->  theta in [0, 2*pi)
// ---------------------------------------------------------------------------
__global__ void k_cas_table(float* __restrict__ T) {
    int idx = blockIdx.x * blockDim.x + threadIdx.x;  // 0 .. 64*16384-1
    int k = idx >> 14;
    int n = idx & (LEN - 1);
    int r = (k * n) & (LEN - 1);
    float th = (float)r * (6.28318530717958647692f / (float)LEN);
    float s, c;
    __sincosf(th, &s, &c);
    T[idx] = c + s;
}

// ---------------------------------------------------------------------------
// Kernel 2: forward DHT (64 modes), split-K GEMM via v_wmma_f32_16x16x4_f32
// wave -> (rowblock of 16 rows, k-chunk of 512); computes all 4 mode tiles
// (64 modes) reusing the A fragment.  Partial sums -> part[s][row][mode].
// ---------------------------------------------------------------------------
__global__ void k_dht_fwd(const float* __restrict__ x, const float* __restrict__ T,
                          float* __restrict__ part) {
    int wave = (blockIdx.x * blockDim.x + threadIdx.x) >> 5;
    int lane = threadIdx.x & 31;
    int rowblock = wave >> 5;       // 0..63
    int s        = wave & 31;       // 0..31  (k-chunk)
    int kbase = s * KCHUNK;
    int lm = lane & 15;
    int lh = lane >> 4;             // 0 or 1

    // A: lane l, vgpr v -> x[rowblock*16 + lm][kbase + 4t + 2*lh + v]
    const float* xrow = x + (size_t)(rowblock * 16 + lm) * LEN + kbase + 2 * lh;
    // B: lane l, vgpr v -> T[mb*16 + lm][kbase + 4t + 2*lh + v]
    const float* trow = T + (size_t)lm * LEN + kbase + 2 * lh;

    v8f c0 = {}, c1 = {}, c2 = {}, c3 = {};
    for (int t = 0; t < KCHUNK / 4; ++t) {
        v2f a  = *(const v2f*)(xrow + 4 * t);
        v2f b0 = *(const v2f*)(trow + 4 * t);
        v2f b1 = *(const v2f*)(trow + (size_t)16 * LEN + 4 * t);
        v2f b2 = *(const v2f*)(trow + (size_t)32 * LEN + 4 * t);
        v2f b3 = *(const v2f*)(trow + (size_t)48 * LEN + 4 * t);
        c0 = __builtin_amdgcn_wmma_f32_16x16x4_f32(false, a, false, b0, (short)0, c0, false, false);
        c1 = __builtin_amdgcn_wmma_f32_16x16x4_f32(false, a, false, b1, (short)0, c1, false, false);
        c2 = __builtin_amdgcn_wmma_f32_16x16x4_f32(false, a, false, b2, (short)0, c2, false, false);
        c3 = __builtin_amdgcn_wmma_f32_16x16x4_f32(false, a, false, b3, (short)0, c3, false, false);
    }

    // C layout: vgpr v -> row = rowblock*16 + v + 8*lh, col = mb*16 + lm
    float* p = part + (size_t)s * (ROWS * MODES);
    int row0 = rowblock * 16 + 8 * lh;
#pragma unroll
    for (int v = 0; v < 8; ++v) {
        p[(size_t)(row0 + v) * MODES +  0 + lm] = c0[v];
        p[(size_t)(row0 + v) * MODES + 16 + lm] = c1[v];
        p[(size_t)(row0 + v) * MODES + 32 + lm] = c2[v];
        p[(size_t)(row0 + v) * MODES + 48 + lm] = c3[v];
    }
}

// ---------------------------------------------------------------------------
// Kernel 3: reduce split-K partials -> xm[row][mode]
// ---------------------------------------------------------------------------
__global__ void k_reduce(const float* __restrict__ part, float* __restrict__ xm) {
    int idx = blockIdx.x * blockDim.x + threadIdx.x;  // 0..65535
    float acc = 0.f;
#pragma unroll
    for (int s = 0; s < KSPLIT; ++s) acc += part[(size_t)s * (ROWS * MODES) + idx];
    xm[idx] = acc;
}

// ---------------------------------------------------------------------------
// Kernel 4: even/odd channel mixing + sin(phase) scale.
// z[b,o,k] = sum_i xm[b,i,k]*ye[i,o,k] + xm[b,i,kf]*yo[i,o,k];  kf = (64-k)&63
// ye/yo = 0.5*(w +/- w_flip);  z *= sin(phase[0,o,k])
// ---------------------------------------------------------------------------
__global__ void k_mix(const float* __restrict__ xm, const float* __restrict__ w1,
                      const float* __restrict__ pw, float* __restrict__ z) {
    int idx = blockIdx.x * blockDim.x + threadIdx.x;  // b*2048 + o*64 + k
    int k = idx & 63;
    int o = (idx >> 6) & 31;
    int b = idx >> 11;
    int kf = (MODES - k) & (MODES - 1);
    float acc = 0.f;
    for (int i = 0; i < CIN; ++i) {
        float xv  = xm[(size_t)(b * CIN + i) * MODES + k];
        float xvf = xm[(size_t)(b * CIN + i) * MODES + kf];
        float w   = w1[(size_t)(i * COUT + o) * MODES + k];
        float wf  = w1[(size_t)(i * COUT + o) * MODES + kf];
        float ye = 0.5f * (w + wf);
        float yo = 0.5f * (w - wf);
        acc = fmaf(xv, ye, fmaf(xvf, yo, acc));
    }
    z[idx] = acc * sinf(pw[(size_t)o * MODES + k]);
}

// ---------------------------------------------------------------------------
// Kernel 5: inverse DHT.  out[row,n] = (1/L) * sum_{k<64} z[row,k]*T[k][n]
// wave -> (time-block of 16, row-group of 8 rowblocks). The full 64x16 cas
// B-tile is held in 16 v2f fragments (32 VGPRs) and reused for 8 row tiles.
// ---------------------------------------------------------------------------
__global__ void k_dht_inv(const float* __restrict__ z, const float* __restrict__ T,
                          float* __restrict__ out) {
    int wave = (blockIdx.x * blockDim.x + threadIdx.x) >> 5;
    int lane = threadIdx.x & 31;
    int nblock = wave >> 3;         // 0..1023 (time tile)
    int rgrp   = wave & 7;          // 0..7
    int lm = lane & 15;
    int lh = lane >> 4;
    int nbase = nblock * 16;

    // B: lane l, vgpr v -> K = 4t + 2*lh + v, N = lm  => T[K][nbase+lm]
    v2f bf[16];
    const float* tp = T + (size_t)(2 * lh) * LEN + nbase + lm;
#pragma unroll
    for (int t = 0; t < 16; ++t) {
        bf[t].x = tp[(size_t)(4 * t) * LEN];
        bf[t].y = tp[(size_t)(4 * t + 1) * LEN];
    }

    const float invL = 1.0f / (float)LEN;
    for (int rb = rgrp * 8; rb < rgrp * 8 + 8; ++rb) {
        v8f c = {};
        const float* zp = z + (size_t)(rb * 16 + lm) * MODES + 2 * lh;
#pragma unroll
        for (int t = 0; t < 16; ++t) {
            v2f a = *(const v2f*)(zp + 4 * t);
            c = __builtin_amdgcn_wmma_f32_16x16x4_f32(false, a, false, bf[t], (short)0, c, false, false);
        }
        int row0 = rb * 16 + 8 * lh;
        float* op = out + (size_t)row0 * LEN + nbase + lm;
#pragma unroll
        for (int v = 0; v < 8; ++v) op[(size_t)v * LEN] = c[v] * invL;
    }
}

// ---------------------------------------------------------------------------
// Launcher. Workspace layout (bytes):
//   [0,        4 MB)  cas table T[64][16384]
//   [4 MB,    12 MB)  split-K partials part[32][1024][64]
//   [12 MB, 12.25 MB) xm[1024][64]
//   [12.25, 12.5 MB)  z[1024][64]
// ---------------------------------------------------------------------------
extern "C" void kernel_launch(void* const* d_in, const int* in_sizes, int n_in,
                              void* d_out, int out_size, void* d_ws, size_t ws_size,
                              hipStream_t stream) {
    (void)in_sizes; (void)n_in; (void)out_size; (void)ws_size;
    const float* x  = (const float*)d_in[0];
    const float* w1 = (const float*)d_in[1];
    const float* pw = (const float*)d_in[2];
    float* out = (float*)d_out;

    char* ws = (char*)d_ws;
    float* T    = (float*)ws;                                        // 4 MB
    float* part = (float*)(ws + (size_t)MODES * LEN * sizeof(float));// 8 MB
    float* xm   = (float*)(ws + (size_t)12 * 1024 * 1024);
    float* z    = xm + (size_t)ROWS * MODES;

    k_cas_table<<<(MODES * LEN) / 256, 256, 0, stream>>>(T);
    k_dht_fwd  <<<(64 * KSPLIT) / 4, 128, 0, stream>>>(x, T, part);   // 2048 waves
    k_reduce   <<<(ROWS * MODES) / 256, 256, 0, stream>>>(part, xm);
    k_mix      <<<(OROWS * MODES) / 256, 256, 0, stream>>>(xm, w1, pw, z);
    k_dht_inv  <<<(1024 * 8) / 4, 128, 0, stream>>>(z, T, out);       // 8192 waves
}